// FlowAlignedSmoothingEffect_56092272885913
// MI455X (gfx1250) — compile-verified
//
#include <hip/hip_runtime.h>
#include <stdint.h>

#define IMG_W 1024
#define IMG_H 1024
#define NCH   3
#define TILE  32
#define HALO  12
#define TW    (TILE + 2*HALO)   /* 56 */
#define NT    (TW*TW)           /* 3136 */

typedef uint32_t u32x4 __attribute__((ext_vector_type(4)));
typedef int      i32x8 __attribute__((ext_vector_type(8)));
typedef int      i32x4 __attribute__((ext_vector_type(4)));

// Issue one 2D TDM tile load: TWxTW f32 tile from global (row stride in elems) into LDS.
// D# per CDNA5 ISA ch.8: group0 = count/lds_addr/global_addr/type, group1 = dims/strides.
__device__ __forceinline__ void tdm_load_tile(uint32_t lds_off, const float* gsrc,
                                              uint32_t row_stride_elems) {
  const uint64_t ga = (uint64_t)(uintptr_t)gsrc;
  u32x4 g0;
  g0[0] = 1u;                                               // count=1, user mode
  g0[1] = lds_off;                                          // LDS byte address
  g0[2] = (uint32_t)ga;                                     // global_addr[31:0]
  g0[3] = (uint32_t)((ga >> 32) & 0x1FFFFFFull) | (2u << 30); // addr[56:32] | type=2

  i32x8 g1;
  const uint32_t td0 = TW, td1 = TW;                        // tensor dims == tile dims (in-bounds by construction)
  const uint64_t s0  = (uint64_t)row_stride_elems;          // tensor_dim0_stride (elems)
  g1[0] = (int)(2u << 16);                                  // wg_mask=0, data_size=2 (4B)
  g1[1] = (int)((td0 & 0xFFFFu) << 16);                     // tensor_dim0[15:0] @ bit48
  g1[2] = (int)((td0 >> 16) | ((td1 & 0xFFFFu) << 16));     // td0[31:16] | td1[15:0]
  g1[3] = (int)((td1 >> 16) | ((uint32_t)TW << 16));        // td1[31:16] | tile_dim0
  g1[4] = (int)(uint32_t)TW;                                // tile_dim1 | tile_dim2=0 (2D)
  g1[5] = (int)(uint32_t)(s0 & 0xFFFFFFFFu);                // stride0[31:0]
  g1[6] = (int)(uint32_t)((s0 >> 32) & 0xFFFFu);            // stride0[47:32] | stride1[15:0]=0
  g1[7] = 0;                                                // stride1[47:16]=0

  i32x4 z4; z4[0] = 0; z4[1] = 0; z4[2] = 0; z4[3] = 0;     // groups 2/3 unused (2D)
  i32x8 z8; z8[0] = 0; z8[1] = 0; z8[2] = 0; z8[3] = 0;
            z8[4] = 0; z8[5] = 0; z8[6] = 0; z8[7] = 0;
  __builtin_amdgcn_tensor_load_to_lds(g0, g1, z4, z4, z8, 0);
}

__global__ __launch_bounds__(256)
void flow_smooth_kernel(const float* __restrict__ x, const float* __restrict__ tg,
                        const float* __restrict__ sigma, float* __restrict__ out) {
  __shared__ float shx[NCH][NT];   // image tile, planar
  __shared__ float sht[2][NT];     // tangent tile, planar

  const int tx0 = blockIdx.x * TILE;
  const int ty0 = blockIdx.y * TILE;
  const int b   = blockIdx.z;

  int ox = tx0 - HALO; ox = (ox < 0) ? 0 : ox; ox = (ox > IMG_W - TW) ? (IMG_W - TW) : ox;
  int oy = ty0 - HALO; oy = (oy < 0) ? 0 : oy; oy = (oy > IMG_H - TW) ? (IMG_H - TW) : oy;

  // Wave 0 stages all 5 planes via the Tensor Data Mover, waits on TENSORcnt,
  // then the workgroup barrier publishes the LDS tiles to the other 7 waves.
  if (threadIdx.x < 32) {
    const size_t plane = (size_t)IMG_H * IMG_W;
    const float* xb = x  + (size_t)b * NCH * plane + (size_t)oy * IMG_W + ox;
    const float* tb = tg + (size_t)b * 2   * plane + (size_t)oy * IMG_W + ox;
    tdm_load_tile((uint32_t)(uintptr_t)&shx[0][0], xb + 0 * plane, IMG_W);
    tdm_load_tile((uint32_t)(uintptr_t)&shx[1][0], xb + 1 * plane, IMG_W);
    tdm_load_tile((uint32_t)(uintptr_t)&shx[2][0], xb + 2 * plane, IMG_W);
    tdm_load_tile((uint32_t)(uintptr_t)&sht[0][0], tb + 0 * plane, IMG_W);
    tdm_load_tile((uint32_t)(uintptr_t)&sht[1][0], tb + 1 * plane, IMG_W);
    __builtin_amdgcn_s_wait_tensorcnt(0);
  }
  __syncthreads();

  // Per-batch uniforms (factor = min(H,W)/1024 = 1 here).
  const float factor     = (float)((IMG_H < IMG_W ? IMG_H : IMG_W)) / 1024.0f;
  const float sg         = sigma[b] * factor;
  const float half_width = 2.0f * sg;
  const float inv2s2     = 1.0f / (2.0f * sg * sg);
  const float stp        = 1.0f / (0.3333f * factor);
  const float invW       = 1.0f / (float)IMG_W;
  const float invH       = 1.0f / (float)IMG_H;

  const int lx  = threadIdx.x & 31;
  const int lyb = threadIdx.x >> 5;

  const float* X0 = &shx[0][0];
  const float* X1 = &shx[1][0];
  const float* X2 = &shx[2][0];
  const float* T0 = &sht[0][0];
  const float* T1 = &sht[1][0];

  #pragma unroll
  for (int sub = 0; sub < 4; ++sub) {
    const int px = tx0 + lx;
    const int py = ty0 + lyb + sub * 8;
    const int tidx = (py - oy) * TW + (px - ox);

    const float p0x = ((float)px + 0.5f) * invW;
    const float p0y = ((float)py + 0.5f) * invH;
    const float t0x = T0[tidx];
    const float t0y = T1[tidx];

    float c0 = 0.f, c1 = 0.f, c2 = 0.f, ss = 0.f;

    #pragma unroll
    for (int dir = 0; dir < 2; ++dir) {
      float vx = dir ? -t0x : t0x;
      float vy = dir ? -t0y : t0y;
      float pxn = p0x + vx * invW;
      float pyn = p0y + vy * invH;
      float r = stp;
      // Steps with r >= half_width contribute k==0 and r is monotone -> early exit.
      for (int it = 0; it < 8 && r < half_width; ++it) {
        float k = __expf(-r * r * inv2s2);
        const bool inb = (pxn >= 0.f) && (pxn < 1.f) && (pyn >= 0.f) && (pyn < 1.f);
        if (!inb) k = 0.f;

        // Bilinear taps (clamp-to-edge in global coords; guaranteed inside tile).
        const float fx = pxn * (float)IMG_W - 0.5f;
        const float fy = pyn * (float)IMG_H - 0.5f;
        const float fx0 = floorf(fx), fy0 = floorf(fy);
        const float wx = fx - fx0,    wy = fy - fy0;
        int x0 = (int)fx0, y0 = (int)fy0;
        x0 = min(max(x0, 0), IMG_W - 1);
        y0 = min(max(y0, 0), IMG_H - 1);
        const int x1i = min(x0 + 1, IMG_W - 1);
        const int y1i = min(y0 + 1, IMG_H - 1);
        const int a0 = min(max(x0  - ox, 0), TW - 1);
        const int a1 = min(max(x1i - ox, 0), TW - 1);
        const int b0 = min(max(y0  - oy, 0), TW - 1);
        const int b1 = min(max(y1i - oy, 0), TW - 1);
        const int i00 = b0 * TW + a0, i01 = b0 * TW + a1;
        const int i10 = b1 * TW + a0, i11 = b1 * TW + a1;
        const float w00 = (1.f - wx) * (1.f - wy), w01 = wx * (1.f - wy);
        const float w10 = (1.f - wx) * wy,         w11 = wx * wy;

        c0 += k * (X0[i00]*w00 + X0[i01]*w01 + X0[i10]*w10 + X0[i11]*w11);
        c1 += k * (X1[i00]*w00 + X1[i01]*w01 + X1[i10]*w10 + X1[i11]*w11);
        c2 += k * (X2[i00]*w00 + X2[i01]*w01 + X2[i10]*w10 + X2[i11]*w11);
        ss += k;

        float tfx = T0[i00]*w00 + T0[i01]*w01 + T0[i10]*w10 + T0[i11]*w11;
        float tfy = T1[i00]*w00 + T1[i01]*w01 + T1[i10]*w10 + T1[i11]*w11;
        const float vt = vx * tfx + vy * tfy;
        if (vt < 0.f) { tfx = -tfx; tfy = -tfy; }
        vx = tfx; vy = tfy;
        pxn += tfx * invW;
        pyn += tfy * invH;
        r += stp;
      }
    }

    const float inv_den = 1.0f / (1.0f + ss);
    const size_t plane = (size_t)IMG_H * IMG_W;
    const size_t obase = (size_t)b * NCH * plane + (size_t)py * IMG_W + (size_t)px;
    out[obase + 0 * plane] = (X0[tidx] + c0) * inv_den;
    out[obase + 1 * plane] = (X1[tidx] + c1) * inv_den;
    out[obase + 2 * plane] = (X2[tidx] + c2) * inv_den;
  }
}

extern "C" void kernel_launch(void* const* d_in, const int* in_sizes, int n_in,
                              void* d_out, int out_size, void* d_ws, size_t ws_size,
                              hipStream_t stream) {
  const float* x  = (const float*)d_in[0];
  const float* tg = (const float*)d_in[1];
  const float* sg = (const float*)d_in[2];
  float* out = (float*)d_out;
  const int B = in_sizes[2];  // sigma has B elements
  dim3 grid(IMG_W / TILE, IMG_H / TILE, B);
  flow_smooth_kernel<<<grid, dim3(256), 0, stream>>>(x, tg, sg, out);
}